// DeepFM_67989332296027
// MI455X (gfx1250) — compile-verified
//
#include <hip/hip_runtime.h>
#include <hip/hip_bf16.h>
#include <math.h>

// ---------------- problem constants ----------------
#define BATCH 16384
#define NF    26
#define NV    100000
#define NE    8
#define ND    16
#define FE    208      // NF*NE
#define NH1   128
#define NH2   64
#define BNEPS 1e-5f

typedef float v2f __attribute__((ext_vector_type(2)));
typedef float v8f __attribute__((ext_vector_type(8)));

// ======================================================================
// Kernel 0: zero the BN statistic accumulators (sum/sumsq for both layers)
// ======================================================================
__global__ void k_init_stats(float* st) {
    int i = blockIdx.x * blockDim.x + threadIdx.x;
    if (i < 2 * NH1 + 2 * NH2) st[i] = 0.0f;
}

// ======================================================================
// Kernel 1: per-row feature construction
//   X[b, j] = (order1 + order2) + o2_flat[b, j] + relu(dense[b]·w_dl[:,j] + b_dl[j])
// one 256-thread block per row
// ======================================================================
__global__ void k_features(const int* __restrict__ sparse,
                           const float* __restrict__ dense,
                           const float* __restrict__ emb1,
                           const float* __restrict__ emb2,
                           const float* __restrict__ w_d1,
                           const float* __restrict__ b_d1,
                           const float* __restrict__ w_dl,
                           const float* __restrict__ b_dl,
                           float* __restrict__ X) {
    __shared__ float o2s[FE];     // gathered order-2 embeddings, flattened
    __shared__ float e1s[NF];     // order-1 scalars
    __shared__ float qs[NF];      // per-field sum of squares
    __shared__ float drow[ND];    // dense row
    __shared__ float sume[NE];    // sum over fields per embedding dim
    __shared__ float sv[2];       // [0]=o1+q combined scalar temp, [1]=row scalar s

    const int b   = blockIdx.x;
    const int tid = threadIdx.x;

    if (tid < NF) {
        const int f   = tid;
        const int idx = sparse[b * NF + f];
        const size_t base = (size_t)f * NV + (size_t)idx;
        e1s[f] = emb1[base];
        const float* p = emb2 + base * NE;
        float q = 0.0f;
        #pragma unroll
        for (int e = 0; e < NE; ++e) {
            float v = p[e];
            o2s[f * NE + e] = v;
            q += v * v;
        }
        qs[f] = q;
    } else if (tid >= 32 && tid < 32 + ND) {
        drow[tid - 32] = dense[(size_t)b * ND + (tid - 32)];
    }
    __syncthreads();

    if (tid < NE) {
        float s = 0.0f;
        for (int f = 0; f < NF; ++f) s += o2s[f * NE + tid];
        sume[tid] = s;
    } else if (tid == NE) {
        float o1 = 0.0f, q = 0.0f;
        for (int f = 0; f < NF; ++f) { o1 += e1s[f]; q += qs[f]; }
        sv[0] = o1 - 0.5f * q;                 // fold -0.5*sumsq into scalar
    }
    __syncthreads();

    if (tid == 0) {
        float ss = 0.0f;
        #pragma unroll
        for (int e = 0; e < NE; ++e) ss += sume[e] * sume[e];
        float s = sv[0] + 0.5f * ss;           // order1_sparse + order2
        #pragma unroll
        for (int k = 0; k < ND; ++k) s = fmaf(drow[k], w_d1[k], s);
        sv[1] = s + b_d1[0];
    }
    __syncthreads();

    if (tid < FE) {
        const int j = tid;
        float acc = b_dl[j];
        #pragma unroll
        for (int k = 0; k < ND; ++k) acc = fmaf(drow[k], w_dl[k * FE + j], acc);
        X[(size_t)b * FE + j] = sv[1] + o2s[j] + fmaxf(acc, 0.0f);
    }
}

// ======================================================================
// Kernel 2: GEMM1  Y1 = X[16384,208] @ w1[208,128] + b1
//   8 waves / block, each wave -> one 16x16 tile (M-tile 16, full N=128)
//   f32 WMMA 16x16x4; fused per-column sum / sumsq atomics for BatchNorm.
// ======================================================================
#define LDA1 (FE + 1)   // 209, odd stride -> conflict-free LDS
__global__ void k_gemm1(const float* __restrict__ X,
                        const float* __restrict__ W1,
                        const float* __restrict__ b1,
                        float* __restrict__ Y1,
                        float* __restrict__ colsum,
                        float* __restrict__ colsq) {
    __shared__ float Xs[16 * LDA1];

    const int tid  = threadIdx.x;
    const int m0   = blockIdx.x * 16;
    const int wave = tid >> 5;
    const int lane = tid & 31;
    const int l16  = lane & 15;
    const int klo  = (lane >> 4) * 2;
    const int n0   = wave * 16;

    // stage A tile (16 x 208) into LDS, coalesced
    for (int i = tid; i < 16 * FE; i += 256) {
        const int r = i / FE, c = i - r * FE;
        Xs[r * LDA1 + c] = X[(size_t)(m0 + r) * FE + c];
    }
    __syncthreads();

    v8f acc = {};
    #pragma unroll 4
    for (int k = 0; k < FE; k += 4) {
        v2f a, b;
        a.x = Xs[l16 * LDA1 + k + klo];
        a.y = Xs[l16 * LDA1 + k + klo + 1];
        b.x = W1[(size_t)(k + klo) * NH1 + n0 + l16];
        b.y = W1[(size_t)(k + klo + 1) * NH1 + n0 + l16];
        acc = __builtin_amdgcn_wmma_f32_16x16x4_f32(false, a, false, b,
                                                    (short)0, acc, false, false);
    }

    // epilogue: + bias, store, fused BN batch-statistic partials
    const int   n  = n0 + l16;
    const float bb = b1[n];
    float ps = 0.0f, pq = 0.0f;
    #pragma unroll
    for (int r = 0; r < 8; ++r) {
        float y = acc[r] + bb;
        const int m = m0 + r + 8 * (lane >> 4);
        Y1[(size_t)m * NH1 + n] = y;
        ps += y;
        pq += y * y;
    }
    ps += __shfl_down(ps, 16, 32);
    pq += __shfl_down(pq, 16, 32);
    if (lane < 16) {
        atomicAdd(&colsum[n], ps);
        atomicAdd(&colsq[n], pq);
    }
}

// ======================================================================
// Kernel 3/5: BN finalize -> affine a,c so that BN+ReLU == relu(a*x + c)
// ======================================================================
__global__ void k_bn_finalize(const float* __restrict__ colsum,
                              const float* __restrict__ colsq,
                              const float* __restrict__ g,
                              const float* __restrict__ be,
                              float* __restrict__ a,
                              float* __restrict__ c,
                              int n, float invB) {
    int i = blockIdx.x * blockDim.x + threadIdx.x;
    if (i < n) {
        float mu  = colsum[i] * invB;
        float var = colsq[i] * invB - mu * mu;
        float s   = g[i] * rsqrtf(var + BNEPS);
        a[i] = s;
        c[i] = be[i] - mu * s;
    }
}

// ======================================================================
// Kernel 4: GEMM2  Y2 = relu(a1*Y1 + c1)[16384,128] @ w2[128,64] + b2
//   BN+ReLU applied while staging tile into LDS; stats fused for layer 2.
// ======================================================================
#define LDA2 (NH1 + 1)  // 129
__global__ void k_gemm2(const float* __restrict__ Y1,
                        const float* __restrict__ a1,
                        const float* __restrict__ c1,
                        const float* __restrict__ W2,
                        const float* __restrict__ b2,
                        float* __restrict__ Y2,
                        float* __restrict__ colsum,
                        float* __restrict__ colsq) {
    __shared__ float Zs[16 * LDA2];

    const int tid  = threadIdx.x;       // 128 threads = 4 waves
    const int m0   = blockIdx.x * 16;
    const int wave = tid >> 5;
    const int lane = tid & 31;
    const int l16  = lane & 15;
    const int klo  = (lane >> 4) * 2;
    const int n0   = wave * 16;

    for (int i = tid; i < 16 * NH1; i += 128) {
        const int r = i >> 7, c = i & (NH1 - 1);
        float y = Y1[(size_t)(m0 + r) * NH1 + c];
        Zs[r * LDA2 + c] = fmaxf(fmaf(a1[c], y, c1[c]), 0.0f);
    }
    __syncthreads();

    v8f acc = {};
    #pragma unroll 4
    for (int k = 0; k < NH1; k += 4) {
        v2f a, b;
        a.x = Zs[l16 * LDA2 + k + klo];
        a.y = Zs[l16 * LDA2 + k + klo + 1];
        b.x = W2[(size_t)(k + klo) * NH2 + n0 + l16];
        b.y = W2[(size_t)(k + klo + 1) * NH2 + n0 + l16];
        acc = __builtin_amdgcn_wmma_f32_16x16x4_f32(false, a, false, b,
                                                    (short)0, acc, false, false);
    }

    const int   n  = n0 + l16;
    const float bb = b2[n];
    float ps = 0.0f, pq = 0.0f;
    #pragma unroll
    for (int r = 0; r < 8; ++r) {
        float y = acc[r] + bb;
        const int m = m0 + r + 8 * (lane >> 4);
        Y2[(size_t)m * NH2 + n] = y;
        ps += y;
        pq += y * y;
    }
    ps += __shfl_down(ps, 16, 32);
    pq += __shfl_down(pq, 16, 32);
    if (lane < 16) {
        atomicAdd(&colsum[n], ps);
        atomicAdd(&colsq[n], pq);
    }
}

// ======================================================================
// Kernel 6: out[b] = b_out + sum_c relu(a2*Y2[b,c] + c2[c]) * w_out[c]
//   one wave per row, lane handles columns (lane, lane+32)
// ======================================================================
__global__ void k_out(const float* __restrict__ Y2,
                      const float* __restrict__ a2,
                      const float* __restrict__ c2,
                      const float* __restrict__ w_out,
                      const float* __restrict__ b_out,
                      float* __restrict__ out) {
    const int lane = threadIdx.x & 31;
    const int row  = blockIdx.x * 8 + (threadIdx.x >> 5);
    const float* yr = Y2 + (size_t)row * NH2;

    float v = fmaxf(fmaf(a2[lane], yr[lane], c2[lane]), 0.0f) * w_out[lane];
    const int c2i = lane + 32;
    v += fmaxf(fmaf(a2[c2i], yr[c2i], c2[c2i]), 0.0f) * w_out[c2i];

    #pragma unroll
    for (int off = 16; off > 0; off >>= 1) v += __shfl_down(v, off, 32);
    if (lane == 0) out[row] = v + b_out[0];
}

// ======================================================================
// host-side launcher
// ======================================================================
extern "C" void kernel_launch(void* const* d_in, const int* in_sizes, int n_in,
                              void* d_out, int out_size, void* d_ws, size_t ws_size,
                              hipStream_t stream) {
    const int*   sparse = (const int*)  d_in[0];
    const float* dense  = (const float*)d_in[1];
    const float* emb1   = (const float*)d_in[2];
    const float* emb2   = (const float*)d_in[3];
    const float* w_d1   = (const float*)d_in[4];
    const float* b_d1   = (const float*)d_in[5];
    const float* w_dl   = (const float*)d_in[6];
    const float* b_dl   = (const float*)d_in[7];
    const float* w1     = (const float*)d_in[8];
    const float* b1     = (const float*)d_in[9];
    const float* g1     = (const float*)d_in[10];
    const float* be1    = (const float*)d_in[11];
    const float* w2     = (const float*)d_in[12];
    const float* b2     = (const float*)d_in[13];
    const float* g2     = (const float*)d_in[14];
    const float* be2    = (const float*)d_in[15];
    const float* w_out  = (const float*)d_in[16];
    const float* b_out  = (const float*)d_in[17];
    float* out = (float*)d_out;

    // workspace layout (floats)
    float* X    = (float*)d_ws;                       // [BATCH, 208]
    float* Y1   = X  + (size_t)BATCH * FE;            // [BATCH, 128]
    float* Y2   = Y1 + (size_t)BATCH * NH1;           // [BATCH, 64]
    float* st   = Y2 + (size_t)BATCH * NH2;           // stats block
    float* sum1 = st;                                  // [128]
    float* sq1  = sum1 + NH1;                          // [128]
    float* sum2 = sq1  + NH1;                          // [64]
    float* sq2  = sum2 + NH2;                          // [64]
    float* a1   = sq2  + NH2;                          // [128]
    float* c1   = a1   + NH1;                          // [128]
    float* a2   = c1   + NH1;                          // [64]
    float* c2   = a2   + NH2;                          // [64]

    const float invB = 1.0f / (float)BATCH;

    k_init_stats<<<1, 512, 0, stream>>>(st);

    k_features<<<BATCH, 256, 0, stream>>>(sparse, dense, emb1, emb2,
                                          w_d1, b_d1, w_dl, b_dl, X);

    k_gemm1<<<BATCH / 16, 256, 0, stream>>>(X, w1, b1, Y1, sum1, sq1);

    k_bn_finalize<<<1, 128, 0, stream>>>(sum1, sq1, g1, be1, a1, c1, NH1, invB);

    k_gemm2<<<BATCH / 16, 128, 0, stream>>>(Y1, a1, c1, w2, b2, Y2, sum2, sq2);

    k_bn_finalize<<<1, 64, 0, stream>>>(sum2, sq2, g2, be2, a2, c2, NH2, invB);

    k_out<<<BATCH / 8, 256, 0, stream>>>(Y2, a2, c2, w_out, b_out, out);
}